// GCN_layer_14491219657224
// MI455X (gfx1250) — compile-verified
//
#include <hip/hip_runtime.h>
#include <hip/hip_bf16.h>

// ---------------------------------------------------------------------------
// GCN layer for MI455X (gfx1250, wave32):
//   h   = x @ W^T + bias          -> f32 WMMA (V_WMMA_F32_16X16X4_F32)
//   out = segment_sum(v * h[col]) -> wave-per-edge + global_atomic_add_f32
// ---------------------------------------------------------------------------

typedef __attribute__((ext_vector_type(2))) float v2f;
typedef __attribute__((ext_vector_type(8))) float v8f;

#define D 128            // D_IN == D_OUT == 128
#define NTILES 8         // 128 output cols / 16

// Each wave computes a 16-row x 128-col tile of h.
// A (16x4 f32) layout per ISA: lanes 0-15 -> M=lane, VGPR0/1 = K0/K1;
//                              lanes 16-31 -> M=lane-16, VGPR0/1 = K2/K3.
// B (4x16 f32) layout: VGPR0: lanes 0-15 = (K0, N=lane), lanes 16-31 = (K2, N=lane-16);
//                      VGPR1: K1 / K3.  => same (row, k+kHi) float2 load shape as A.
// C/D (16x16 f32): VGPR j: lanes 0-15 -> (M=j, N=lane), lanes 16-31 -> (M=8+j, N=lane-16).
__global__ void __launch_bounds__(256)
gcn_gemm_wmma(const float* __restrict__ x,
              const float* __restrict__ w,      // [128,128] row-major, h[n,o]=sum_i x[n,i]*w[o,i]
              const float* __restrict__ bias,   // [128]
              float* __restrict__ h,            // [N,128]
              int N)
{
    const int lane   = threadIdx.x & 31;
    const int waveId = threadIdx.x >> 5;
    const int m0     = blockIdx.x * 128 + waveId * 16;
    if (m0 >= N) return;                        // wave-uniform exit; EXEC stays all-1s

    const int nIdx = lane & 15;                 // N / M-within-16 coordinate
    const int kHi  = (lane >> 4) * 2;           // 0 for lanes 0-15, 2 for lanes 16-31

    // Clamp tail rows for A loads (results masked at store time).
    int rowA = m0 + nIdx;
    if (rowA >= N) rowA = N - 1;
    const float* aBase = x + (size_t)rowA * D + kHi;

    // Init accumulators with bias: every C element held by this lane sits in
    // column (t*16 + nIdx), so all 8 VGPRs of tile t start at bias[t*16+nIdx].
    v8f acc[NTILES];
#pragma unroll
    for (int t = 0; t < NTILES; ++t) {
        const float b = bias[t * 16 + nIdx];
        v8f c;
#pragma unroll
        for (int j = 0; j < 8; ++j) c[j] = b;
        acc[t] = c;
    }

    // K loop: 32 steps of K=4.  One A load feeds 8 independent WMMA chains.
    for (int k0 = 0; k0 < D; k0 += 4) {
        const v2f a = *(const v2f*)(aBase + k0);
#pragma unroll
        for (int t = 0; t < NTILES; ++t) {
            const v2f b = *(const v2f*)(w + (size_t)(t * 16 + nIdx) * D + k0 + kHi);
            acc[t] = __builtin_amdgcn_wmma_f32_16x16x4_f32(
                /*neg_a=*/false, a, /*neg_b=*/false, b,
                /*c_mod=*/(short)0, acc[t], /*reuse_a=*/false, /*reuse_b=*/false);
        }
    }

    // Store C tiles (row-guarded for the grid tail).
    const int mBase = m0 + ((lane >> 4) << 3);  // +8 for high half-wave
#pragma unroll
    for (int t = 0; t < NTILES; ++t) {
#pragma unroll
        for (int j = 0; j < 8; ++j) {
            const int row = mBase + j;
            if (row < N) h[(size_t)row * D + t * 16 + nIdx] = acc[t][j];
        }
    }
}

// One wave32 per edge: coalesced 512B gather of h[col] (float4 per lane),
// scale by edge value, 4 fire-and-forget f32 atomics per lane into out[row].
// h and out are L2-resident (51.2 MB each vs 192 MB L2).
__global__ void __launch_bounds__(256)
gcn_scatter(const float* __restrict__ h,
            const float* __restrict__ vals,
            const int*   __restrict__ rows,
            const int*   __restrict__ cols,
            float* __restrict__ out,
            int E)
{
    const int edge = (int)((blockIdx.x * (unsigned)blockDim.x + threadIdx.x) >> 5);
    const int lane = threadIdx.x & 31;
    if (edge >= E) return;

    const int   r = rows[edge];
    const int   c = cols[edge];
    const float v = vals[edge];

    const float4 m = ((const float4*)(h + (size_t)c * D))[lane];
    float* o = out + (size_t)r * D + lane * 4;
    atomicAdd(o + 0, v * m.x);
    atomicAdd(o + 1, v * m.y);
    atomicAdd(o + 2, v * m.z);
    atomicAdd(o + 3, v * m.w);
}

extern "C" void kernel_launch(void* const* d_in, const int* in_sizes, int n_in,
                              void* d_out, int out_size, void* d_ws, size_t ws_size,
                              hipStream_t stream)
{
    const float* x    = (const float*)d_in[0];
    const float* w    = (const float*)d_in[1];
    const float* bias = (const float*)d_in[2];
    const float* ev   = (const float*)d_in[3];
    const int*   er   = (const int*)d_in[4];
    const int*   ec   = (const int*)d_in[5];
    float* out = (float*)d_out;
    float* h   = (float*)d_ws;                  // N*128 f32 = 51.2 MB scratch

    const int N = in_sizes[0] / D;              // 100000
    const int E = in_sizes[3];                  // 3200000

    // 1) h = x W^T + b  (8 waves/block, 16 rows/wave -> 128 rows/block)
    const int gblocks = (N + 127) / 128;
    gcn_gemm_wmma<<<gblocks, 256, 0, stream>>>(x, w, bias, h, N);

    // 2) out = 0
    hipMemsetAsync(d_out, 0, (size_t)out_size * sizeof(float), stream);

    // 3) out[r] += v * h[c]  (8 edges/block)
    const int eblocks = (E + 7) / 8;
    gcn_scatter<<<eblocks, 256, 0, stream>>>(h, ev, er, ec, out, E);
}